// EmbeddingModel_3779571220787
// MI455X (gfx1250) — compile-verified
//
#include <hip/hip_runtime.h>

typedef __attribute__((ext_vector_type(2))) float v2f;
typedef __attribute__((ext_vector_type(8))) float v8f;

#define EMBED   128
#define N_POS   10
#define N_NEG   50
#define N_CTX   60          // 10 pos + 50 neg context rows per batch item
#define TILE_M  16
#define N_TILES 4           // 4*16 = 64 rows >= 60 (rows 60..63 never loaded, masked out)
#define ROW_DW  132         // padded LDS row stride (dwords): A-frag b64 reads hit all 64 banks
#define WPB     4           // waves per block (128 threads)

__device__ __forceinline__ float log_sigmoid(float x) {
    // numerically stable: min(x,0) - log(1 + exp(-|x|))
    float ax = __builtin_fabsf(x);
    return fminf(x, 0.0f) - __logf(1.0f + __expf(-ax));
}

// CDNA5 async gather: copy 512B (32 lanes x b128) global -> LDS, tracked by ASYNCcnt.
__device__ __forceinline__ void async_row_b128(const float* gaddr, unsigned lds_off) {
    asm volatile("global_load_async_to_lds_b128 %0, %1, off"
                 :: "v"(lds_off), "v"((unsigned long long)(uintptr_t)gaddr)
                 : "memory");
}
__device__ __forceinline__ void wait_async0() {
    asm volatile("s_wait_asynccnt 0x0" ::: "memory");
}
__device__ __forceinline__ void wait_ds0() {
    asm volatile("s_wait_dscnt 0x0" ::: "memory");
}

__global__ __launch_bounds__(WPB * 32, 1)
void w2v_neg_sampling_loss_kernel(const int*   __restrict__ input_labels,  // [B,1]
                                  const int*   __restrict__ pos_labels,    // [B,10]
                                  const int*   __restrict__ neg_labels,    // [B,50]
                                  const float* __restrict__ in_embed,      // [V,128]
                                  const float* __restrict__ out_embed,     // [V,128]
                                  float*       __restrict__ out,           // [B]
                                  int batch)
{
    __shared__ float smem[WPB][TILE_M * ROW_DW + EMBED];

    const int wave = threadIdx.x >> 5;
    const int lane = threadIdx.x & 31;
    const int b    = blockIdx.x * WPB + wave;     // uniform per wave
    if (b >= batch) return;                       // wave-uniform exit

    float* ctx  = &smem[wave][0];                 // 16 rows x ROW_DW
    float* vrow = &smem[wave][TILE_M * ROW_DW];   // center vector, 128 f32

    // low 32 bits of a flat shared pointer == wave-relative LDS byte offset
    const unsigned ctx_off  = (unsigned)(uintptr_t)ctx;
    const unsigned vrow_off = (unsigned)(uintptr_t)vrow;
    const unsigned lane_b   = (unsigned)lane * 16u;   // 16B per lane within a 512B row

    // ---- async-stage center vector v = in_embed[input_labels[b]] ----
    {
        long long vlab = (long long)input_labels[b];
        async_row_b128(in_embed + vlab * EMBED + lane * 4, vrow_off + lane_b);
    }

    // WMMA f32 16x16x4 A-fragment mapping (ISA 7.12.2):
    //   lanes 0-15 : M = lane,    VGPR0=K(k0+0), VGPR1=K(k0+1)
    //   lanes 16-31: M = lane-16, VGPR0=K(k0+2), VGPR1=K(k0+3)
    const int m  = lane & 15;
    const int kh = (lane >> 4) << 1;              // 0 (low half) or 2 (high half)

    float partial = 0.0f;

    for (int t = 0; t < N_TILES; ++t) {
        // previous tile's ds_loads must drain before the async engine refills the buffer
        wait_ds0();

        // ---- issue async gathers for this tile's context rows (512B each) ----
        for (int r = 0; r < TILE_M; ++r) {
            int i = t * TILE_M + r;               // uniform
            if (i < N_CTX) {
                long long lab = (i < N_POS)
                    ? (long long)pos_labels[b * N_POS + i]
                    : (long long)neg_labels[b * N_NEG + (i - N_POS)];
                async_row_b128(out_embed + lab * EMBED + lane * 4,
                               ctx_off + (unsigned)(r * ROW_DW) * 4u + lane_b);
            }
        }

        // ---- prefetch next tile's rows into cache while this tile computes ----
        if (t + 1 < N_TILES) {
            for (int r = 0; r < TILE_M; ++r) {
                int i = (t + 1) * TILE_M + r;     // uniform
                if (i < N_CTX) {
                    long long lab = (i < N_POS)
                        ? (long long)pos_labels[b * N_POS + i]
                        : (long long)neg_labels[b * N_NEG + (i - N_POS)];
                    __builtin_prefetch(out_embed + lab * EMBED + lane * 4, 0, 3);
                }
            }
        }

        wait_async0();                            // tile (and v row, t==0) resident in LDS

        // ---- K-reduction via V_WMMA_F32_16X16X4_F32, accumulate 16 dots ----
        v8f acc = {};
        const float* arow = ctx + m * ROW_DW + kh;
        #pragma unroll 8
        for (int k0 = 0; k0 < EMBED; k0 += 4) {
            v2f a  = *(const v2f*)(arow + k0);            // A: ctx rows, bank-conflict-free
            v2f bb = *(const v2f*)(vrow + k0 + kh);       // B: v broadcast to all 16 cols
            acc = __builtin_amdgcn_wmma_f32_16x16x4_f32(
                      /*neg_a=*/false, a, /*neg_b=*/false, bb,
                      /*c_mod=*/(short)0, acc, /*reuse_a=*/false, /*reuse_b=*/false);
        }

        // ---- D layout: VGPR r holds row (t*16 + r) on lanes 0-15, row (t*16+8+r) on 16-31;
        //      every column identical (B replicated); mask removes unloaded rows (NaN-safe). ----
        int base = t * TILE_M + ((lane >> 4) << 3);
        #pragma unroll
        for (int r = 0; r < 8; ++r) {
            int i = base + r;
            float x = acc[r];
            float term = (i < N_POS) ? log_sigmoid(x) : log_sigmoid(-x);
            partial += (i < N_CTX) ? term : 0.0f;
        }
    }

    // lanes 0-15 hold one identical partial, lanes 16-31 the other; combine halves.
    float total = partial + __shfl_xor(partial, 16, 32);
    if (lane == 0) out[b] = -total;
}

extern "C" void kernel_launch(void* const* d_in, const int* in_sizes, int n_in,
                              void* d_out, int out_size, void* d_ws, size_t ws_size,
                              hipStream_t stream) {
    const int*   input_labels = (const int*)  d_in[0];
    const int*   pos_labels   = (const int*)  d_in[1];
    const int*   neg_labels   = (const int*)  d_in[2];
    const float* in_embed     = (const float*)d_in[3];
    const float* out_embed    = (const float*)d_in[4];
    float*       out          = (float*)      d_out;

    int batch  = in_sizes[0];                      // 16384 (input_labels is [B,1])
    int blocks = (batch + WPB - 1) / WPB;

    hipLaunchKernelGGL(w2v_neg_sampling_loss_kernel,
                       dim3(blocks), dim3(WPB * 32), 0, stream,
                       input_labels, pos_labels, neg_labels,
                       in_embed, out_embed, out, batch);
}